// Bargrain_50251117363882
// MI455X (gfx1250) — compile-verified
//
#include <hip/hip_runtime.h>
#include <hip/hip_bf16.h>

// ---------------------------------------------------------------------------
// CDNA5 / gfx1250 implementation. wave32. Matrix math via V_WMMA_F32_16X16X4_F32
// (exact f32 — the GCN aggregation matrix is data-dependent 0/1·rsqrt products).
// Fragment layouts per CDNA5 ISA §7.12.2:
//   A (16x4 f32): lanes 0-15 hold rows M=0..15; VGPR0=K(0|2), VGPR1=K(1|3),
//                 lanes16-31 take K+2.
//   B (4x16 f32): lanes = N, VGPR0=K(0|2), VGPR1=K(1|3) (symmetric).
//   C/D (16x16):  lanes 0-15 N, VGPR r -> M=r (lanes16-31 -> M=8+r).
// ---------------------------------------------------------------------------

typedef float v2f __attribute__((ext_vector_type(2)));
typedef float v8f __attribute__((ext_vector_type(8)));

#define CDIV(a, b) (((a) + (b) - 1) / (b))

#define LDA_S 132  // 128 + 4: 16B-aligned rows, 4-bank skew
#define LDB_S 68   // 64 + 4

// ---------------------------------------------------------------------------
// Generic f32 WMMA GEMM:  C[m][n] = sum_k op(A)[m][k] * B[k][n]
//   transA=0 : A is M x K row-major (lda = row stride)
//   transA=1 : A is K x M row-major (lda = row stride) i.e. C = A^T B
//   B is K x N row-major (ldb == N), C is M x N row-major.
//   blockIdx.z = batch (strides sA/sB/sC) OR split-K slice (atomic f32 output).
// Block: 256 threads = 8 waves; block tile 128x64; wave grid 4(M) x 2(N);
// each wave owns a 2x2 grid of 16x16 C tiles (4 WMMAs per fragment set).
// Fast path (tile fully in bounds, K%16==0): unguarded float4 global loads,
// pointer-increment addressing, b128 LDS stores, next-chunk global prefetch.
// ---------------------------------------------------------------------------
__global__ __launch_bounds__(256) void k_gemm_wmma(
    const float* __restrict__ A, const float* __restrict__ B,
    float* __restrict__ C, int M, int N, int K, int lda, int transA,
    long long sA, long long sB, long long sC,
    const float* __restrict__ bias, int act, int splitK, int atomicOut) {
  __shared__ float As[16][LDA_S];
  __shared__ float Bs[16][LDB_S];

  const int tid = threadIdx.x;
  const int wave = tid >> 5, lane = tid & 31;
  const int half = lane >> 4, l = lane & 15;
  const int wm = wave & 3, wn = wave >> 2;
  const int tmA = wm * 32;  // wave's 32-row M span (2 tiles)
  const int tnB = wn * 32;  // wave's 32-col N span (2 tiles)
  const int m0 = blockIdx.x * 128, n0 = blockIdx.y * 64;

  int kbeg = 0, kend = K;
  if (splitK > 1) {
    int chunk = (K + splitK - 1) / splitK;
    kbeg = blockIdx.z * chunk;
    kend = kbeg + chunk;
    if (kend > K) kend = K;
  } else {
    long long b = blockIdx.z;
    A += b * sA; B += b * sB; C += b * sC;
  }

  const v8f vzero = {0.f, 0.f, 0.f, 0.f, 0.f, 0.f, 0.f, 0.f};
  v8f acc00 = vzero, acc01 = vzero, acc10 = vzero, acc11 = vzero;

  const bool fast = (m0 + 128 <= M) && (n0 + 64 <= N) &&
                    (((kend - kbeg) & 15) == 0);

  if (fast) {
    // per-thread source pointers, advanced by a constant stride per K-chunk
    const float* a0p;
    const float* a1p;
    size_t stepA;
    int kT = 0, mT = 0, kNT = 0, mNT = 0;
    if (transA) {
      kT = tid >> 5;            // 0..7
      mT = (tid & 31) * 4;      // 0..124
      a0p = A + (size_t)(kbeg + kT) * lda + m0 + mT;
      a1p = a0p + (size_t)8 * lda;   // rows k+8..k+15
      stepA = (size_t)16 * lda;
    } else {
      mNT = tid >> 2;           // 0..63
      kNT = (tid & 3) * 4;      // 0,4,8,12
      a0p = A + (size_t)(m0 + mNT) * lda + kbeg + kNT;
      a1p = a0p + (size_t)64 * lda;  // rows m+64..m+127
      stepA = 16;
    }
    const int kB = tid >> 4;        // 0..15
    const int nB = (tid & 15) * 4;  // 0..60
    const float* bp = B + (size_t)(kbeg + kB) * N + n0 + nB;
    const size_t stepB = (size_t)16 * N;

    const int nch = (kend - kbeg) >> 4;
    for (int c = 0; c < nch; ++c) {
      float4 va0 = *(const float4*)a0p;
      float4 va1 = *(const float4*)a1p;
      float4 vb = *(const float4*)bp;
      if (transA) {
        *(float4*)&As[kT][mT] = va0;
        *(float4*)&As[kT + 8][mT] = va1;
      } else {
        As[kNT + 0][mNT] = va0.x; As[kNT + 1][mNT] = va0.y;
        As[kNT + 2][mNT] = va0.z; As[kNT + 3][mNT] = va0.w;
        As[kNT + 0][mNT + 64] = va1.x; As[kNT + 1][mNT + 64] = va1.y;
        As[kNT + 2][mNT + 64] = va1.z; As[kNT + 3][mNT + 64] = va1.w;
      }
      *(float4*)&Bs[kB][nB] = vb;
      a0p += stepA; a1p += stepA; bp += stepB;
      __builtin_prefetch(a0p, 0, 1);  // global_prefetch_b8, speculative
      __builtin_prefetch(bp, 0, 1);
      __syncthreads();
#pragma unroll
      for (int kk = 0; kk < 16; kk += 4) {
        v2f a0, a1, b0, b1;
        a0.x = As[kk + 2 * half    ][tmA + l];
        a0.y = As[kk + 2 * half + 1][tmA + l];
        a1.x = As[kk + 2 * half    ][tmA + 16 + l];
        a1.y = As[kk + 2 * half + 1][tmA + 16 + l];
        b0.x = Bs[kk + 2 * half    ][tnB + l];
        b0.y = Bs[kk + 2 * half + 1][tnB + l];
        b1.x = Bs[kk + 2 * half    ][tnB + 16 + l];
        b1.y = Bs[kk + 2 * half + 1][tnB + 16 + l];
        acc00 = __builtin_amdgcn_wmma_f32_16x16x4_f32(false, a0, false, b0,
                                                      (short)0, acc00, false, false);
        acc01 = __builtin_amdgcn_wmma_f32_16x16x4_f32(false, a0, false, b1,
                                                      (short)0, acc01, false, false);
        acc10 = __builtin_amdgcn_wmma_f32_16x16x4_f32(false, a1, false, b0,
                                                      (short)0, acc10, false, false);
        acc11 = __builtin_amdgcn_wmma_f32_16x16x4_f32(false, a1, false, b1,
                                                      (short)0, acc11, false, false);
      }
      __syncthreads();
    }
  } else {
    // guarded scalar path (ragged tiles / N=32 head GEMM)
    for (int kc = kbeg; kc < kend; kc += 16) {
#pragma unroll
      for (int r = 0; r < 8; ++r) {  // A: 16 x 128
        int idx = tid + r * 256;
        int k, mm;
        if (transA) { k = idx >> 7; mm = idx & 127; }
        else        { mm = idx >> 4; k = idx & 15; }
        int gk = kc + k, gm = m0 + mm;
        float v = 0.f;
        if (gk < kend && gm < M)
          v = transA ? A[(size_t)gk * lda + gm] : A[(size_t)gm * lda + gk];
        As[k][mm] = v;
      }
#pragma unroll
      for (int r = 0; r < 4; ++r) {  // B: 16 x 64
        int idx = tid + r * 256;
        int k = idx >> 6, nn = idx & 63;
        int gk = kc + k, gn = n0 + nn;
        Bs[k][nn] = (gk < kend && gn < N) ? B[(size_t)gk * N + gn] : 0.f;
      }
      __syncthreads();
#pragma unroll
      for (int kk = 0; kk < 16; kk += 4) {
        v2f a0, a1, b0, b1;
        a0.x = As[kk + 2 * half    ][tmA + l];
        a0.y = As[kk + 2 * half + 1][tmA + l];
        a1.x = As[kk + 2 * half    ][tmA + 16 + l];
        a1.y = As[kk + 2 * half + 1][tmA + 16 + l];
        b0.x = Bs[kk + 2 * half    ][tnB + l];
        b0.y = Bs[kk + 2 * half + 1][tnB + l];
        b1.x = Bs[kk + 2 * half    ][tnB + 16 + l];
        b1.y = Bs[kk + 2 * half + 1][tnB + 16 + l];
        acc00 = __builtin_amdgcn_wmma_f32_16x16x4_f32(false, a0, false, b0,
                                                      (short)0, acc00, false, false);
        acc01 = __builtin_amdgcn_wmma_f32_16x16x4_f32(false, a0, false, b1,
                                                      (short)0, acc01, false, false);
        acc10 = __builtin_amdgcn_wmma_f32_16x16x4_f32(false, a1, false, b0,
                                                      (short)0, acc10, false, false);
        acc11 = __builtin_amdgcn_wmma_f32_16x16x4_f32(false, a1, false, b1,
                                                      (short)0, acc11, false, false);
      }
      __syncthreads();
    }
  }

  // ---- epilogue: 2x2 tiles per wave ----
#pragma unroll
  for (int t = 0; t < 2; ++t) {
#pragma unroll
    for (int u = 0; u < 2; ++u) {
      v8f acc = t ? (u ? acc11 : acc10) : (u ? acc01 : acc00);
      int mb = m0 + tmA + t * 16 + half * 8;
      int n = n0 + tnB + u * 16 + l;
#pragma unroll
      for (int r = 0; r < 8; ++r) {
        int m = mb + r;
        if (m < M && n < N) {
          float v = acc[r];
          if (atomicOut) {
            atomicAdd(&C[(size_t)m * N + n], v);
          } else {
            if (bias) v += bias[n];
            if (act == 1) v = fmaxf(v, 0.f);
            else if (act == 2) v = (v >= 0.f) ? v : 0.2f * v;
            C[(size_t)m * N + n] = v;
          }
        }
      }
    }
  }
}

// ---------------------------------------------------------------------------
// Small helper kernels
// ---------------------------------------------------------------------------
__global__ void k_transpose(const float* __restrict__ in, float* __restrict__ out,
                            int R, int C) {
  __shared__ float tile[32][33];
  int bx = blockIdx.x * 32, by = blockIdx.y * 32;
  int tx = threadIdx.x, ty = threadIdx.y;
#pragma unroll
  for (int j = 0; j < 32; j += 8) {
    int r = by + ty + j, c = bx + tx;
    if (r < R && c < C) tile[ty + j][tx] = in[(size_t)r * C + c];
  }
  __syncthreads();
#pragma unroll
  for (int j = 0; j < 32; j += 8) {
    int r = bx + ty + j, c = by + tx;
    if (r < C && c < R) out[(size_t)r * R + c] = tile[tx][ty + j];
  }
}

__global__ void k_fill(float* __restrict__ p, float v, int n) {
  int i = blockIdx.x * blockDim.x + threadIdx.x;
  if (i < n) p[i] = v;
}

__global__ void k_relu(float* __restrict__ p, int n) {
  int i = blockIdx.x * blockDim.x + threadIdx.x;
  if (i < n) p[i] = fmaxf(p[i], 0.f);
}

// ls/lr projections of relu(t):  ls[n][s] = sum_k relu(t[n][k]) * Wcat[s][k]
__global__ __launch_bounds__(256) void k_lslr(const float* __restrict__ t,
                                              const float* __restrict__ wcat,
                                              float* __restrict__ ls,
                                              float* __restrict__ lr) {
  __shared__ float sd[4][256];
  int n = blockIdx.x, k = threadIdx.x;
  float u = fmaxf(t[(size_t)n * 256 + k], 0.f);
  sd[0][k] = u * wcat[k];              // Ws row 0 (cols 0..255)
  sd[1][k] = u * wcat[512 + k];        // Ws row 1
  sd[2][k] = u * wcat[256 + k];        // Wr row 0 (cols 256..511)
  sd[3][k] = u * wcat[512 + 256 + k];  // Wr row 1
  __syncthreads();
  for (int s = 128; s > 0; s >>= 1) {
    if (k < s) {
      sd[0][k] += sd[0][k + s]; sd[1][k] += sd[1][k + s];
      sd[2][k] += sd[2][k + s]; sd[3][k] += sd[3][k + s];
    }
    __syncthreads();
  }
  if (k == 0) {
    ls[n * 2] = sd[0][0]; ls[n * 2 + 1] = sd[1][0];
    lr[n * 2] = sd[2][0]; lr[n * 2 + 1] = sd[3][0];
  }
}

__device__ __forceinline__ float gumbelize(float u) {
  u = fminf(fmaxf(u, 1e-9f), 1.f - 1e-9f);
  return -logf(-logf(u));
}

// Dense-GCN normalization: deg[j] = 1 + sum_i A[i][j];  disD = rsqrt(deg)
__global__ __launch_bounds__(512) void k_deg_dense(
    const float* __restrict__ gum, const float* __restrict__ ls,
    const float* __restrict__ lr, const float* __restrict__ fcb,
    float* __restrict__ disD) {
  __shared__ float lr0[512], lr1[512];
  int b = blockIdx.x, j = threadIdx.x;
  int m = b * 512 + j;
  lr0[j] = lr[m * 2]; lr1[j] = lr[m * 2 + 1];
  __syncthreads();
  float ls0 = ls[m * 2] + fcb[0], ls1 = ls[m * 2 + 1] + fcb[1];
  const float2* g2 = (const float2*)gum + (size_t)b * 512 * 512;
  float acc = 0.f;
  for (int i = 0; i < 512; ++i) {
    float2 g = g2[(size_t)i * 512 + j];
    float y0 = lr0[i] + ls0 + gumbelize(g.x);
    float y1 = lr1[i] + ls1 + gumbelize(g.y);
    acc += (y0 >= y1) ? 1.f : 0.f;  // argmax ties -> index 0
  }
  disD[m] = rsqrtf(acc + 1.f);
}

// Q[b][j][i] = M[j][i] = disD[i]*disD[j]*(A[j][i] + (i==j))  (transposed store)
__global__ __launch_bounds__(512) void k_build_q(
    const float* __restrict__ gum, const float* __restrict__ ls,
    const float* __restrict__ lr, const float* __restrict__ fcb,
    const float* __restrict__ disD, float* __restrict__ Q) {
  int bj = blockIdx.x;  // b*512 + j
  int b = bj >> 9, j = bj & 511;
  int i = threadIdx.x;
  int nj = b * 512 + j, ni = b * 512 + i;
  float l0 = lr[nj * 2] + fcb[0], l1 = lr[nj * 2 + 1] + fcb[1];
  float2 g = ((const float2*)gum)[(size_t)bj * 512 + i];
  float y0 = l0 + ls[ni * 2] + gumbelize(g.x);
  float y1 = l1 + ls[ni * 2 + 1] + gumbelize(g.y);
  float a = ((y0 >= y1) ? 1.f : 0.f) + ((i == j) ? 1.f : 0.f);
  Q[(size_t)bj * 512 + i] = disD[ni] * disD[nj] * a;
}

// Sparse in-degree accumulation
__global__ void k_degcount(const int* __restrict__ col, float* __restrict__ deg,
                           int E) {
  int i = blockIdx.x * blockDim.x + threadIdx.x;
  if (i < E) atomicAdd(&deg[col[i]], 1.f);
}

__global__ void k_rsqrt(const float* __restrict__ deg, float* __restrict__ dis,
                        int n) {
  int i = blockIdx.x * blockDim.x + threadIdx.x;
  if (i < n) { float d = deg[i]; dis[i] = (d > 0.f) ? rsqrtf(d) : 0.f; }
}

// out[n][f] = xw[n][f]*dis[n]^2 + bias[f]  (self-loop term + bias, overwrites)
__global__ void k_selfinit(const float* __restrict__ xw,
                           const float* __restrict__ dis,
                           const float* __restrict__ bias,
                           float* __restrict__ out, int F) {
  int n = blockIdx.x, f = threadIdx.x;
  float d = dis[n];
  out[(size_t)n * F + f] = xw[(size_t)n * F + f] * d * d + bias[f];
}

// One block per edge: out[col] += xw[row] * dis[row]*dis[col]
__global__ void k_scatter(const int* __restrict__ row, const int* __restrict__ col,
                          const float* __restrict__ dis,
                          const float* __restrict__ xw, float* __restrict__ out,
                          int F) {
  int e = blockIdx.x;
  int r = row[e], c = col[e];
  float nrm = dis[r] * dis[c];
  int f = threadIdx.x;
  atomicAdd(&out[(size_t)c * F + f], xw[(size_t)r * F + f] * nrm);
}

// Et[(n*256 + k)][b]: interleaved embedding transposed to K x batch (GEMM-ready)
__global__ __launch_bounds__(256) void k_build_et(const float* __restrict__ xop,
                                                  const float* __restrict__ xc,
                                                  float* __restrict__ Et) {
  __shared__ float tile[256][33];
  int n = blockIdx.x, f = threadIdx.x;  // n = node-in-sample (0..511)
  for (int b = 0; b < 32; ++b) {
    float v = (f < 128) ? xop[((size_t)b * 512 + n) * 128 + f]
                        : xc[((size_t)b * 512 + n) * 128 + (f - 128)];
    tile[f][b] = v;
  }
  __syncthreads();
#pragma unroll
  for (int r = 0; r < 32; ++r) {
    int idx = threadIdx.x + r * 256;
    int k = idx >> 5, b = idx & 31;
    Et[((size_t)n * 256 + k) * 32 + b] = tile[k][b];
  }
}

// fcn1 epilogue + fcn2 + fcn3, single LDS-resident block.
__global__ __launch_bounds__(256) void k_fcn_tail(
    const float* __restrict__ cpre, const float* __restrict__ b1,
    const float* __restrict__ W2, const float* __restrict__ b2,
    const float* __restrict__ W3, const float* __restrict__ b3,
    float* __restrict__ out) {
  __shared__ float h1s[32 * 256];
  __shared__ float h2s[32 * 32];
  int tid = threadIdx.x;
#pragma unroll
  for (int r = 0; r < 32; ++r) {
    int idx = tid + r * 256;
    int b = idx >> 8, o = idx & 255;
    float v = cpre[o * 32 + b] + b1[o];  // cpre is (256 x 32) = h1^T
    h1s[b * 256 + o] = (v >= 0.f) ? v : 0.2f * v;
  }
  __syncthreads();
#pragma unroll
  for (int r = 0; r < 4; ++r) {
    int idx = tid + r * 256;
    int b = idx >> 5, j = idx & 31;
    float s = 0.f;
    for (int o = 0; o < 256; ++o) s += h1s[b * 256 + o] * W2[j * 256 + o];
    s += b2[j];
    h2s[b * 32 + j] = (s >= 0.f) ? s : 0.2f * s;
  }
  __syncthreads();
  if (tid < 64) {
    int b = tid >> 1, c = tid & 1;
    float s = 0.f;
    for (int j = 0; j < 32; ++j) s += h2s[b * 32 + j] * W3[c * 32 + j];
    out[b * 2 + c] = s + b3[c];
  }
}

// ---------------------------------------------------------------------------
extern "C" void kernel_launch(void* const* d_in, const int* in_sizes, int n_in,
                              void* d_out, int out_size, void* d_ws,
                              size_t ws_size, hipStream_t stream) {
  (void)in_sizes; (void)n_in; (void)out_size; (void)ws_size;
  const float* x   = (const float*)d_in[0];
  const float* t   = (const float*)d_in[1];
  const float* gum = (const float*)d_in[2];
  const int*   ei  = (const int*)d_in[3];
  const float* w1  = (const float*)d_in[4];
  const float* b1  = (const float*)d_in[5];
  const float* w2  = (const float*)d_in[6];
  const float* b2  = (const float*)d_in[7];
  const float* wc  = (const float*)d_in[8];
  const float* bc  = (const float*)d_in[9];
  const float* f1w = (const float*)d_in[10];
  const float* f1b = (const float*)d_in[11];
  const float* f2w = (const float*)d_in[12];
  const float* f2b = (const float*)d_in[13];
  const float* f3w = (const float*)d_in[14];
  const float* f3b = (const float*)d_in[15];
  float* out = (float*)d_out;

  const int Nn = 512, BZ = 32, NTOT = 16384, E = 524288, F1 = 256, F2 = 128;
  const int* row = ei;
  const int* col = ei + E;

  // ---- workspace carve (~90 MB, fully rewritten every call) ----
  char* base = (char*)d_ws;
  size_t o = 0;
  auto carve = [&](size_t bytes) -> float* {
    float* r = (float*)(base + o);
    o += (bytes + 255) & ~(size_t)255;
    return r;
  };
  float* xw1  = carve((size_t)NTOT * F1 * 4);       // 16 MB, later reused as Et
  float* Q    = carve((size_t)BZ * Nn * Nn * 4);    // 32 MB
  float* h    = carve((size_t)NTOT * F1 * 4);       // 16 MB, later out1
  float* hw2  = carve((size_t)NTOT * F2 * 4);       // 8 MB, later xw2c
  float* xop  = carve((size_t)NTOT * F2 * 4);       // 8 MB
  float* out2 = carve((size_t)NTOT * F2 * 4);       // 8 MB
  float* ls   = carve((size_t)NTOT * 2 * 4);
  float* lr   = carve((size_t)NTOT * 2 * 4);
  float* disD = carve((size_t)NTOT * 4);
  float* w1t  = carve((size_t)512 * 256 * 4);
  float* w2t  = carve((size_t)256 * 128 * 4);
  float* degS = carve((size_t)NTOT * 4);
  float* disS = carve((size_t)NTOT * 4);
  float* cpre = carve((size_t)256 * 32 * 4);
  float* Et   = xw1;  // alias: xw1 dead after sparse layer-1 scatter
  float* out1 = h;    // alias: h dead after hw2 GEMM
  float* xw2c = hw2;  // alias: hw2 dead after dense layer-2 GEMM

  dim3 blk(256);

  // ---- prep: weight transposes (K-major B for WMMA), ls/lr ----
  k_transpose<<<dim3(CDIV(512, 32), CDIV(256, 32)), dim3(32, 8), 0, stream>>>(
      w1, w1t, 256, 512);
  k_transpose<<<dim3(CDIV(256, 32), CDIV(128, 32)), dim3(32, 8), 0, stream>>>(
      w2, w2t, 128, 256);
  k_lslr<<<NTOT, 256, 0, stream>>>(t, wc, ls, lr);

  // ---- dense path: adjacency stats + normalized aggregation matrix ----
  k_deg_dense<<<BZ, 512, 0, stream>>>(gum, ls, lr, bc, disD);
  k_build_q<<<NTOT, 512, 0, stream>>>(gum, ls, lr, bc, disD, Q);

  // xw1 = x @ W1^T             (16384x512 * 512x256)
  k_gemm_wmma<<<dim3(CDIV(NTOT, 128), CDIV(F1, 64), 1), blk, 0, stream>>>(
      x, w1t, xw1, NTOT, F1, 512, 512, 0, 0, 0, 0, nullptr, 0, 1, 0);
  // h = relu(Q^T @ xw1 + b1)   batched over 32 samples
  k_gemm_wmma<<<dim3(4, 4, BZ), blk, 0, stream>>>(
      Q, xw1, h, 512, F1, 512, 512, 1, (long long)512 * 512,
      (long long)512 * F1, (long long)512 * F1, b1, 1, 1, 0);
  // hw2 = h @ W2^T
  k_gemm_wmma<<<dim3(CDIV(NTOT, 128), CDIV(F2, 64), 1), blk, 0, stream>>>(
      h, w2t, hw2, NTOT, F2, F1, F1, 0, 0, 0, 0, nullptr, 0, 1, 0);
  // x_op = Q^T @ hw2 + b2      batched
  k_gemm_wmma<<<dim3(4, 2, BZ), blk, 0, stream>>>(
      Q, hw2, xop, 512, F2, 512, 512, 1, (long long)512 * 512,
      (long long)512 * F2, (long long)512 * F2, b2, 0, 1, 0);

  // ---- sparse path ----
  k_fill<<<CDIV(NTOT, 256), 256, 0, stream>>>(degS, 1.f, NTOT);  // self loops
  k_degcount<<<CDIV(E, 256), 256, 0, stream>>>(col, degS, E);
  k_rsqrt<<<CDIV(NTOT, 256), 256, 0, stream>>>(degS, disS, NTOT);
  k_selfinit<<<NTOT, F1, 0, stream>>>(xw1, disS, b1, out1, F1);
  k_scatter<<<E, F1, 0, stream>>>(row, col, disS, xw1, out1, F1);
  k_relu<<<CDIV(NTOT * F1, 256), 256, 0, stream>>>(out1, NTOT * F1);
  // xw2c = x_corr1 @ W2^T
  k_gemm_wmma<<<dim3(CDIV(NTOT, 128), CDIV(F2, 64), 1), blk, 0, stream>>>(
      out1, w2t, xw2c, NTOT, F2, F1, F1, 0, 0, 0, 0, nullptr, 0, 1, 0);
  k_selfinit<<<NTOT, F2, 0, stream>>>(xw2c, disS, b2, out2, F2);
  k_scatter<<<E, F2, 0, stream>>>(row, col, disS, xw2c, out2, F2);

  // ---- MLP head ----
  // Et[k][b] (131072 x 32) from interleaved [x_op | x_corr2]
  k_build_et<<<512, 256, 0, stream>>>(xop, out2, Et);
  k_fill<<<CDIV(256 * 32, 256), 256, 0, stream>>>(cpre, 0.f, 256 * 32);
  // cpre(256x32) += fcn1_w(256x131072) @ Et  — split-K=128, atomic f32 reduce
  k_gemm_wmma<<<dim3(CDIV(256, 128), CDIV(32, 64), 128), blk, 0, stream>>>(
      f1w, Et, cpre, 256, 32, 131072, 131072, 0, 0, 0, 0, nullptr, 0, 128, 1);
  // leaky(fcn1) -> fcn2 -> fcn3
  k_fcn_tail<<<1, 256, 0, stream>>>(cpre, f1b, f2w, f2b, f3w, f3b, out);
}